// AttentionalAggregation_34505767256374
// MI455X (gfx1250) — compile-verified
//
#include <hip/hip_runtime.h>
#include <hip/hip_bf16.h>
#include <math.h>

#define N_SEG   50000
#define M_ROWS  800000
#define D_DIM   128
#define OUT_DIM 128
#define K_DIM   256   // 2*D

typedef __attribute__((ext_vector_type(2))) float v2f;
typedef __attribute__((ext_vector_type(8))) float v8f;

// ---------------------------------------------------------------------------
// Kernel 1: segment boundaries from sorted segment ids.
// ---------------------------------------------------------------------------
__global__ void seg_bounds_kernel(const int* __restrict__ seg,
                                  int* __restrict__ seg_start,
                                  int* __restrict__ seg_end) {
    int i = blockIdx.x * blockDim.x + threadIdx.x;
    if (i >= M_ROWS) return;
    int s = seg[i];
    if (i == 0 || seg[i - 1] != s) seg_start[s] = i;
    if (i == M_ROWS - 1 || seg[i + 1] != s) seg_end[s] = i + 1;
}

// ---------------------------------------------------------------------------
// Kernel 2: one wave32 per segment; lane owns 4 contiguous columns (float4).
// Streams 410 MB of lane_encoding exactly once -> HBM-bandwidth bound.
// pooled[seg, 0:128] = max, pooled[seg, 128:256] = mean.
// ---------------------------------------------------------------------------
__global__ void seg_reduce_kernel(const float* __restrict__ lane_enc,
                                  const int* __restrict__ seg_start,
                                  const int* __restrict__ seg_end,
                                  float* __restrict__ pooled) {
    int wave = (int)((blockIdx.x * blockDim.x + threadIdx.x) >> 5);  // grid sized exactly
    int lane = threadIdx.x & 31;

    int s0 = seg_start[wave];
    int s1 = seg_end[wave];

    const float* p = lane_enc + (size_t)s0 * D_DIM + lane * 4;
    float4 mx = make_float4(-INFINITY, -INFINITY, -INFINITY, -INFINITY);
    float4 sm = make_float4(0.f, 0.f, 0.f, 0.f);

    for (int r = s0; r < s1; ++r) {
        float4 v = *(const float4*)p;
        mx.x = fmaxf(mx.x, v.x); mx.y = fmaxf(mx.y, v.y);
        mx.z = fmaxf(mx.z, v.z); mx.w = fmaxf(mx.w, v.w);
        sm.x += v.x; sm.y += v.y; sm.z += v.z; sm.w += v.w;
        p += D_DIM;
    }

    float inv = 1.0f / (float)(s1 - s0);
    float4 mean = make_float4(sm.x * inv, sm.y * inv, sm.z * inv, sm.w * inv);

    float* orow = pooled + (size_t)wave * K_DIM + lane * 4;
    *(float4*)(orow)         = mx;    // first 128 cols: segment max
    *(float4*)(orow + D_DIM) = mean;  // next 128 cols: segment mean
}

// ---------------------------------------------------------------------------
// Kernel 3: out = relu(pooled @ W.T + b) via V_WMMA_F32_16X16X4_F32.
// One wave per 16-row strip (blockDim=32 so EXEC is always all-ones).
// A fragment (16x4): lane = M-row, VGPR0/1 = K{0,1} (lanes 0-15) or K{2,3}
// (lanes 16-31)  -> single aligned 8-byte load.
// B fragment (4x16): lane = N-col (B[k][n] = W[n][k]), same K striping.
// C/D: VGPR r -> M=r (lanes 0-15), M=r+8 (lanes 16-31); lane -> N.
// ---------------------------------------------------------------------------
__global__ void gemm_relu_kernel(const float* __restrict__ pooled,
                                 const float* __restrict__ Wm,
                                 const float* __restrict__ bias,
                                 float* __restrict__ out) {
    int wave = blockIdx.x;          // 3125 strips, one wave each
    int lane = threadIdx.x & 31;
    int half = lane >> 4;           // 0: K{0,1}, 1: K{2,3}
    int l15  = lane & 15;

    int m0 = wave * 16;
    const float* arow = pooled + (size_t)(m0 + l15) * K_DIM + 2 * half;
    const float* wrow = Wm     + (size_t)l15        * K_DIM + 2 * half;

    v8f acc[8];
#pragma unroll
    for (int nt = 0; nt < 8; ++nt) acc[nt] = (v8f){0.f,0.f,0.f,0.f,0.f,0.f,0.f,0.f};

    for (int k = 0; k < K_DIM; k += 4) {
        v2f a = *(const v2f*)(arow + k);
#pragma unroll
        for (int nt = 0; nt < 8; ++nt) {
            v2f bfrag = *(const v2f*)(wrow + (size_t)nt * 16 * K_DIM + k);
            acc[nt] = __builtin_amdgcn_wmma_f32_16x16x4_f32(
                /*neg_a=*/false, a, /*neg_b=*/false, bfrag,
                /*c_mod=*/(short)0, acc[nt], /*reuse_a=*/false, /*reuse_b=*/false);
        }
    }

#pragma unroll
    for (int nt = 0; nt < 8; ++nt) {
        int n = nt * 16 + l15;
        float bb = bias[n];
#pragma unroll
        for (int r = 0; r < 8; ++r) {
            int m = m0 + r + 8 * half;
            float v = acc[nt][r] + bb;
            out[(size_t)m * OUT_DIM + n] = fmaxf(v, 0.0f);
        }
    }
}

// ---------------------------------------------------------------------------
// Launch: inputs are {obs_encoding, lane_encoding, same_obs_mask, W, b}.
// obs_encoding only provides N and is unused. Workspace layout:
//   [0, 51.2MB)      pooled (50000 x 256 f32)
//   [51.2MB, +200KB) seg_start (int32)
//   then             seg_end   (int32)
// ---------------------------------------------------------------------------
extern "C" void kernel_launch(void* const* d_in, const int* in_sizes, int n_in,
                              void* d_out, int out_size, void* d_ws, size_t ws_size,
                              hipStream_t stream) {
    const float* lane_enc = (const float*)d_in[1];
    const int*   seg      = (const int*)d_in[2];
    const float* Wm       = (const float*)d_in[3];
    const float* bias     = (const float*)d_in[4];
    float*       out      = (float*)d_out;

    float* pooled   = (float*)d_ws;
    int*   segstart = (int*)((char*)d_ws + (size_t)N_SEG * K_DIM * sizeof(float));
    int*   segend   = segstart + N_SEG;

    // 800000 / 256 = 3125 blocks exactly
    seg_bounds_kernel<<<(M_ROWS + 255) / 256, 256, 0, stream>>>(seg, segstart, segend);

    // 50000 segments, 8 waves (256 threads) per block -> 6250 blocks exactly
    seg_reduce_kernel<<<N_SEG / 8, 256, 0, stream>>>(lane_enc, segstart, segend, pooled);

    // 50000 / 16 = 3125 strips, one wave32 per block (EXEC all-ones for WMMA)
    gemm_relu_kernel<<<N_SEG / 16, 32, 0, stream>>>(pooled, Wm, bias, out);
}